// RWGCN_84576495993316
// MI455X (gfx1250) — compile-verified
//
#include <hip/hip_runtime.h>
#include <math.h>

#define NNODES 50000
#define NREL   8
#define HID    128
#define NCLS   32
#define KHOP   3
#define NEDGE  800000

typedef float v2f __attribute__((ext_vector_type(2)));
typedef float v8f __attribute__((ext_vector_type(8)));

// ---------------------------------------------------------------------------
// f32 WMMA: D(16x16) += A(16x4) * B(4x16), fp32 throughout (matches reference
// precision; workload is scatter-bound so there is no win from bf16/fp8).
// Lane layouts per CDNA5 ISA 7.12.2:
//   A: lane l holds row m=(l&15), K = k0 + 2*(l>>4) + {0,1} in VGPR {0,1}
//   B: lane l holds col n=(l&15), K = k0 + 2*(l>>4) + {0,1} in VGPR {0,1}
//   C/D: lane l, vgpr j -> (m = j + 8*(l>>4), n = (l&15))
// ---------------------------------------------------------------------------
static __device__ __forceinline__ v8f wmma_f32_acc(v8f c, float a0, float a1,
                                                   float b0, float b1) {
  v2f a; a[0] = a0; a[1] = a1;
  v2f b; b[0] = b0; b[1] = b1;
#if __has_builtin(__builtin_amdgcn_wmma_f32_16x16x4_f32)
  return __builtin_amdgcn_wmma_f32_16x16x4_f32(false, a, false, b,
                                               (short)0, c, false, false);
#else
  v8f d = c;
  asm volatile("v_wmma_f32_16x16x4_f32 %0, %1, %2, %0"
               : "+v"(d) : "v"(a), "v"(b));
  return d;
#endif
}

// ---------------------------------------------------------------------------
// Pass 1: degrees, per-(dst,rel) edge counts, relation histogram
// ---------------------------------------------------------------------------
__global__ void k_deg(const int* __restrict__ src, const int* __restrict__ dst,
                      const int* __restrict__ et, float* degs, float* degd,
                      float* cnt, int* relCnt) {
  int e = blockIdx.x * blockDim.x + threadIdx.x;
  if (e >= NEDGE) return;
  int s = src[e], d = dst[e], t = et[e];
  atomicAdd(&degs[s], 1.0f);
  atomicAdd(&degd[d], 1.0f);
  atomicAdd(&cnt[d * NREL + t], 1.0f);
  atomicAdd(&relCnt[t], 1);
}

__global__ void k_off(const int* __restrict__ relCnt, int* relOff, int* tilePre) {
  relOff[0] = 0; tilePre[0] = 0;
  for (int r = 0; r < NREL; ++r) {
    relOff[r + 1]  = relOff[r] + relCnt[r];
    tilePre[r + 1] = tilePre[r] + (relCnt[r] + 15) / 16;
  }
}

__global__ void k_scatter_rel(const int* __restrict__ et,
                              const int* __restrict__ relOff,
                              int* relCur, int* relEdges) {
  int e = blockIdx.x * blockDim.x + threadIdx.x;
  if (e >= NEDGE) return;
  int t = et[e];
  int pos = atomicAdd(&relCur[t], 1);
  relEdges[relOff[t] + pos] = e;
}

__global__ void k_ew(const int* __restrict__ src, const int* __restrict__ dst,
                     const float* __restrict__ degs, const float* __restrict__ degd,
                     float* __restrict__ ew) {
  int e = blockIdx.x * blockDim.x + threadIdx.x;
  if (e >= NEDGE) return;
  float a = degs[src[e]], b = degd[dst[e]];
  float ia = a > 0.0f ? 1.0f / sqrtf(a) : 0.0f;
  float ib = b > 0.0f ? 1.0f / sqrtf(b) : 0.0f;
  ew[e] = ia * ib;
}

// coeff[k] = softmax(hop_logits)[k] * exp(-exp(log_scale)*k), coeff[4]=sigmoid(gate)
__global__ void k_coeff(const float* __restrict__ log_scale,
                        const float* __restrict__ hl,
                        const float* __restrict__ gate, float* coeff) {
  float m = hl[0];
  for (int k = 1; k <= KHOP; ++k) m = fmaxf(m, hl[k]);
  float e[KHOP + 1], sum = 0.0f;
  for (int k = 0; k <= KHOP; ++k) { e[k] = expf(hl[k] - m); sum += e[k]; }
  float s = expf(log_scale[0]);
  for (int k = 0; k <= KHOP; ++k)
    coeff[k] = (e[k] / sum) * expf(-s * (float)k);
  coeff[4] = 1.0f / (1.0f + expf(-gate[0]));
}

// ---------------------------------------------------------------------------
// Dense node GEMM: out[n,0:O] = A[n,0:128] @ W[128,O] + bias  (f32 WMMA)
// 4 waves / block, 16 rows per wave.  Requires nrows % 16 == 0 (50000 is),
// so no per-row guards -> straight-line coalesced stores.
// ---------------------------------------------------------------------------
__global__ void k_dense_gemm(const float* __restrict__ A, const float* __restrict__ W,
                             const float* __restrict__ bias, float* __restrict__ out,
                             int nrows, int O) {
  const int lane = threadIdx.x & 31;
  const int wid  = threadIdx.x >> 5;
  const int half = lane >> 4;
  const int col  = lane & 15;
  const int row_base = (blockIdx.x * 4 + wid) * 16;
  if (row_base >= nrows) return;
  const float* Arow = A + (size_t)(row_base + col) * HID;
  for (int n0 = 0; n0 < O; n0 += 16) {
    v8f c = {};
#pragma unroll 4
    for (int k0 = 0; k0 < HID; k0 += 4) {
      const int ka = k0 + 2 * half;                       // even -> 8B aligned
      const float2 av = *(const float2*)(Arow + ka);      // global_load_b64
      const float b0 = W[(size_t)ka * O + n0 + col];
      const float b1 = W[(size_t)(ka + 1) * O + n0 + col];
      c = wmma_f32_acc(c, av.x, av.y, b0, b1);
    }
    const float bv = bias[n0 + col];
    float* orow = out + (size_t)(row_base + 8 * half) * O + n0 + col;
#pragma unroll
    for (int j = 0; j < 8; ++j)
      orow[(size_t)j * O] = c[j] + bv;                    // rows row_base+j+8*half
  }
}

// ---------------------------------------------------------------------------
// RGCN edge-tile GEMM: for 16 edges of relation r, D = X[src]@W_r, then
// atomically scatter row i * (1/cnt[dst,r]) into out[dst].  Equivalent to
// mean-per-(dst,rel) then @W_r.
// ---------------------------------------------------------------------------
__global__ void k_edge_gemm(const float* __restrict__ X, const float* __restrict__ W_rel,
                            const float* __restrict__ cnt, const int* __restrict__ relOff,
                            const int* __restrict__ tilePre, const int* __restrict__ relEdges,
                            const int* __restrict__ src, const int* __restrict__ dst,
                            float* __restrict__ out, int O) {
  const int lane = threadIdx.x & 31;
  const int wid  = threadIdx.x >> 5;
  const int half = lane >> 4;
  const int col  = lane & 15;
  const int tile = blockIdx.x * 4 + wid;
  const int total = tilePre[NREL];
  if (tile >= total) return;
  int r = 0;
  while (r < NREL - 1 && tile >= tilePre[r + 1]) ++r;
  const int start  = relOff[r] + (tile - tilePre[r]) * 16;
  const int nvalid = min(16, relOff[r + 1] - start);

  int e_src = 0, e_dst = -1;
  float e_scl = 0.0f;
  if (lane < nvalid) {
    int e  = relEdges[start + lane];
    e_src  = src[e];
    e_dst  = dst[e];
    float cv = cnt[e_dst * NREL + r];
    e_scl  = 1.0f / fmaxf(cv, 1.0f);
  }
  const int arow = __shfl(e_src, col, 32);   // A row for this lane (m = lane&15)
  const float* Arow = X + (size_t)arow * HID;
  const float* Wr   = W_rel + (size_t)r * HID * O;

  for (int n0 = 0; n0 < O; n0 += 16) {
    v8f c = {};
#pragma unroll 4
    for (int k0 = 0; k0 < HID; k0 += 4) {
      const int ka = k0 + 2 * half;
      const float2 av = *(const float2*)(Arow + ka);
      const float b0 = Wr[(size_t)ka * O + n0 + col];
      const float b1 = Wr[(size_t)(ka + 1) * O + n0 + col];
      c = wmma_f32_acc(c, av.x, av.y, b0, b1);
    }
#pragma unroll
    for (int j = 0; j < 8; ++j) {
      int m  = j + 8 * half;
      int d  = __shfl(e_dst, m, 32);
      float sc = __shfl(e_scl, m, 32);
      if (d >= 0) atomicAdd(&out[(size_t)d * O + n0 + col], c[j] * sc);
    }
  }
}

// ---------------------------------------------------------------------------
// Wavelet hop: nxt[dst] += ew * cur[src]; one thread per (edge, 4 floats)
// ---------------------------------------------------------------------------
__global__ void k_hop(const float* __restrict__ cur, const int* __restrict__ src,
                      const int* __restrict__ dst, const float* __restrict__ ew,
                      float* __restrict__ nxt) {
  unsigned tid = blockIdx.x * blockDim.x + threadIdx.x;
  if (tid >= (unsigned)NEDGE * 32u) return;
  int e  = (int)(tid >> 5);
  int ch = ((int)tid & 31) * 4;
  float w = ew[e];
  int s = src[e], d = dst[e];
  const float4 v = *(const float4*)(cur + (size_t)s * HID + ch);
  float* p = nxt + (size_t)d * HID + ch;
  atomicAdd(p + 0, v.x * w);
  atomicAdd(p + 1, v.y * w);
  atomicAdd(p + 2, v.z * w);
  atomicAdd(p + 3, v.w * w);
}

// ---------------------------------------------------------------------------
// Elementwise helpers
// ---------------------------------------------------------------------------
__global__ void k_scale_init(float* __restrict__ wav, const float* __restrict__ x,
                             const float* __restrict__ coeff, int n) {
  int i = blockIdx.x * blockDim.x + threadIdx.x;
  if (i < n) wav[i] = coeff[0] * x[i];
}
__global__ void k_axpy(float* __restrict__ wav, const float* __restrict__ buf,
                       const float* __restrict__ coeff, int k, int n) {
  int i = blockIdx.x * blockDim.x + threadIdx.x;
  if (i < n) wav[i] += coeff[k] * buf[i];
}
__global__ void k_relu_combine(float* __restrict__ r, const float* __restrict__ w,
                               const float* __restrict__ coeff, int n) {
  int i = blockIdx.x * blockDim.x + threadIdx.x;
  if (i < n) {
    float v = r[i] + coeff[4] * w[i];
    r[i] = v > 0.0f ? v : 0.0f;
  }
}
__global__ void k_out_combine(float* __restrict__ out, const float* __restrict__ r,
                              const float* __restrict__ w, const float* __restrict__ coeff,
                              int n) {
  int i = blockIdx.x * blockDim.x + threadIdx.x;
  if (i < n) out[i] = r[i] + coeff[4] * w[i];
}

// ---------------------------------------------------------------------------
extern "C" void kernel_launch(void* const* d_in, const int* in_sizes, int n_in,
                              void* d_out, int out_size, void* d_ws, size_t ws_size,
                              hipStream_t stream) {
  (void)in_sizes; (void)n_in; (void)out_size; (void)ws_size;
  // ---- inputs (setup_inputs order) ----
  const int*   ei       = (const int*)d_in[0];      // [2, E]
  const int*   etype    = (const int*)d_in[1];      // [E]
  const float* node_emb = (const float*)d_in[2];    // [N, 128]
  const float* W_rel1   = (const float*)d_in[3];    // [8,128,128]
  const float* W_root1  = (const float*)d_in[4];    // [128,128]
  const float* b1       = (const float*)d_in[5];
  const float* ls1      = (const float*)d_in[6];
  const float* hl1      = (const float*)d_in[7];
  const float* W_wav1   = (const float*)d_in[8];
  const float* b_wav1   = (const float*)d_in[9];
  const float* W_rel2   = (const float*)d_in[10];   // [8,128,32]
  const float* W_root2  = (const float*)d_in[11];   // [128,32]
  const float* b2       = (const float*)d_in[12];
  const float* ls2      = (const float*)d_in[13];
  const float* hl2      = (const float*)d_in[14];
  const float* W_wav2   = (const float*)d_in[15];
  const float* b_wav2   = (const float*)d_in[16];
  const float* gate1    = (const float*)d_in[17];
  const float* gate2    = (const float*)d_in[18];

  const int* src = ei;
  const int* dst = ei + NEDGE;
  float* out = (float*)d_out;

  // ---- workspace partition ----
  char* p = (char*)d_ws;
  auto take = [&](size_t bytes) -> void* {
    void* q = (void*)p;
    p += (bytes + 255) & ~(size_t)255;
    return q;
  };
  // zero-init region (contiguous -> one memset)
  char*  zbase   = p;
  float* degS    = (float*)take(sizeof(float) * NNODES);
  float* degD    = (float*)take(sizeof(float) * NNODES);
  float* cnt     = (float*)take(sizeof(float) * NNODES * NREL);
  int*   relCnt  = (int*)take(sizeof(int) * NREL);
  int*   relCur  = (int*)take(sizeof(int) * NREL);
  size_t zbytes  = (size_t)(p - zbase);
  // rest
  int*   relOff   = (int*)take(sizeof(int) * (NREL + 1));
  int*   tilePre  = (int*)take(sizeof(int) * (NREL + 1));
  int*   relEdges = (int*)take(sizeof(int) * NEDGE);
  float* ew       = (float*)take(sizeof(float) * NEDGE);
  float* coeff1   = (float*)take(sizeof(float) * 8);
  float* coeff2   = (float*)take(sizeof(float) * 8);
  const int NH = NNODES * HID;
  const int NC = NNODES * NCLS;
  float* bufA = (float*)take(sizeof(float) * NH);   // ping / w1
  float* bufB = (float*)take(sizeof(float) * NH);   // pong
  float* bufC = (float*)take(sizeof(float) * NH);   // wavelet accumulator
  float* bufD = (float*)take(sizeof(float) * NH);   // rgcn1 out -> x1
  float* bufE = (float*)take(sizeof(float) * NC);   // w2
  float* bufF = (float*)take(sizeof(float) * NC);   // rgcn2 out

  // ---- launch geometry ----
  const int TB = 256;
  const int gE   = (NEDGE + TB - 1) / TB;
  const int gNH  = (NH + TB - 1) / TB;
  const int gNC  = (NC + TB - 1) / TB;
  const int gHop = (int)(((unsigned)NEDGE * 32u + TB - 1) / TB);
  const int gGemmN = (NNODES + 63) / 64;                       // dense: 4 waves x 16 rows
  const int maxTiles = (NEDGE + 15) / 16 + NREL;
  const int gEdgeG = (maxTiles + 3) / 4;                       // edge gemm: 4 waves/block

  // ---- graph setup ----
  hipMemsetAsync(zbase, 0, zbytes, stream);
  k_deg<<<gE, TB, 0, stream>>>(src, dst, etype, degS, degD, cnt, relCnt);
  k_off<<<1, 1, 0, stream>>>(relCnt, relOff, tilePre);
  k_scatter_rel<<<gE, TB, 0, stream>>>(etype, relOff, relCur, relEdges);
  k_ew<<<gE, TB, 0, stream>>>(src, dst, degS, degD, ew);

  // ================= layer 1 =================
  k_coeff<<<1, 1, 0, stream>>>(ls1, hl1, gate1, coeff1);
  // r1 = x @ W_root1 + b1   (into bufD), then relation scatter-GEMM adds
  k_dense_gemm<<<gGemmN, 128, 0, stream>>>(node_emb, W_root1, b1, bufD, NNODES, HID);
  k_edge_gemm<<<gEdgeG, 128, 0, stream>>>(node_emb, W_rel1, cnt, relOff, tilePre,
                                          relEdges, src, dst, bufD, HID);
  // wavelet accumulation in bufC; hops ping-pong A/B starting from node_emb
  k_scale_init<<<gNH, TB, 0, stream>>>(bufC, node_emb, coeff1, NH);
  hipMemsetAsync(bufB, 0, sizeof(float) * NH, stream);
  k_hop<<<gHop, TB, 0, stream>>>(node_emb, src, dst, ew, bufB);
  k_axpy<<<gNH, TB, 0, stream>>>(bufC, bufB, coeff1, 1, NH);
  hipMemsetAsync(bufA, 0, sizeof(float) * NH, stream);
  k_hop<<<gHop, TB, 0, stream>>>(bufB, src, dst, ew, bufA);
  k_axpy<<<gNH, TB, 0, stream>>>(bufC, bufA, coeff1, 2, NH);
  hipMemsetAsync(bufB, 0, sizeof(float) * NH, stream);
  k_hop<<<gHop, TB, 0, stream>>>(bufA, src, dst, ew, bufB);
  k_axpy<<<gNH, TB, 0, stream>>>(bufC, bufB, coeff1, 3, NH);
  // w1 = wav @ W_wav1 + b_wav1  (into bufA)
  k_dense_gemm<<<gGemmN, 128, 0, stream>>>(bufC, W_wav1, b_wav1, bufA, NNODES, HID);
  // x1 = relu(r1 + sigmoid(g1)*w1)  (in place in bufD)
  k_relu_combine<<<gNH, TB, 0, stream>>>(bufD, bufA, coeff1, NH);

  // ================= layer 2 =================
  k_coeff<<<1, 1, 0, stream>>>(ls2, hl2, gate2, coeff2);
  k_dense_gemm<<<gGemmN, 128, 0, stream>>>(bufD, W_root2, b2, bufF, NNODES, NCLS);
  k_edge_gemm<<<gEdgeG, 128, 0, stream>>>(bufD, W_rel2, cnt, relOff, tilePre,
                                          relEdges, src, dst, bufF, NCLS);
  k_scale_init<<<gNH, TB, 0, stream>>>(bufC, bufD, coeff2, NH);
  hipMemsetAsync(bufB, 0, sizeof(float) * NH, stream);
  k_hop<<<gHop, TB, 0, stream>>>(bufD, src, dst, ew, bufB);
  k_axpy<<<gNH, TB, 0, stream>>>(bufC, bufB, coeff2, 1, NH);
  hipMemsetAsync(bufA, 0, sizeof(float) * NH, stream);
  k_hop<<<gHop, TB, 0, stream>>>(bufB, src, dst, ew, bufA);
  k_axpy<<<gNH, TB, 0, stream>>>(bufC, bufA, coeff2, 2, NH);
  hipMemsetAsync(bufB, 0, sizeof(float) * NH, stream);
  k_hop<<<gHop, TB, 0, stream>>>(bufA, src, dst, ew, bufB);
  k_axpy<<<gNH, TB, 0, stream>>>(bufC, bufB, coeff2, 3, NH);
  k_dense_gemm<<<gGemmN, 128, 0, stream>>>(bufC, W_wav2, b_wav2, bufE, NNODES, NCLS);
  // out = r2 + sigmoid(g2)*w2
  k_out_combine<<<gNC, TB, 0, stream>>>(out, bufF, bufE, coeff2, NC);
}